// GCN_1211180777896
// MI455X (gfx1250) — compile-verified
//
#include <hip/hip_runtime.h>
#include <hip/hip_bf16.h>

// ---------------- WMMA types (CDNA5 / gfx1250, wave32) ----------------
typedef __attribute__((ext_vector_type(2))) float v2f;
typedef __attribute__((ext_vector_type(8))) float v8f;

#define V_N  25000
#define E_N  150000
#define VP_N 25024   // 391 * 64, padded row count for 64-row WMMA M-blocks

// ---------------- degree / norm ----------------
__global__ void k_deg(const int* __restrict__ src, float* __restrict__ deg, int E) {
    int t = blockIdx.x * blockDim.x + threadIdx.x;
    if (t < E) atomicAdd(&deg[src[t]], 1.0f);
}

__global__ void k_norm(const int* __restrict__ src, const int* __restrict__ dst,
                       const float* __restrict__ deg, float* __restrict__ nrm, int E) {
    int t = blockIdx.x * blockDim.x + threadIdx.x;
    if (t >= E) return;
    float a = deg[src[t]], b = deg[dst[t]];
    float ia = (a > 0.f) ? rsqrtf(a) : 0.f;
    float ib = (b > 0.f) ? rsqrtf(b) : 0.f;
    nrm[t] = ia * ib;
}

// copy mesh template into working mesh + padded feature buffer (pad rows zeroed)
__global__ void k_initmesh(const float* __restrict__ mt, float* __restrict__ mesh,
                           float* __restrict__ meshPad, int V, int VP) {
    int t = blockIdx.x * blockDim.x + threadIdx.x;
    if (t >= VP * 3) return;
    int v = t / 3;
    float val = (v < V) ? mt[t] : 0.f;
    meshPad[t] = val;
    if (v < V) mesh[t] = val;
}

// ---------------- weight pre-pack into WMMA-B layout --------------------------
// Wp[q][n][2] = { W[2q][n], W[2q+1][n] } with K zero-padded; one aligned b64
// per B fetch in the GEMM inner loop, address linear in k.
__global__ void k_packW(const float* __restrict__ Wsrc, float* __restrict__ Wp,
                        int Cin, int Cout, int nQ) {
    int t = blockIdx.x * blockDim.x + threadIdx.x;
    if (t >= nQ * Cout) return;
    int q = t / Cout, n = t - q * Cout;
    float a = (2 * q     < Cin) ? Wsrc[(size_t)(2 * q) * Cout + n]     : 0.f;
    float b = (2 * q + 1 < Cin) ? Wsrc[(size_t)(2 * q + 1) * Cout + n] : 0.f;
    Wp[(size_t)t * 2 + 0] = a;
    Wp[(size_t)t * 2 + 1] = b;
}

// ---------------- sparse part: tx1[dst] -= norm * x[src] ----------------
__global__ void k_scatter(const int* __restrict__ src, const int* __restrict__ dst,
                          const float* __restrict__ nrm,
                          const float* __restrict__ x, int ldx,
                          float* __restrict__ tx, int ldt,
                          int Cin, int E, int nC4) {
    long t = (long)blockIdx.x * blockDim.x + threadIdx.x;
    if (t >= (long)E * nC4) return;
    int e  = (int)(t / nC4);
    int c0 = (int)(t - (long)e * nC4) * 4;
    int s = src[e], d = dst[e];
    float w = -nrm[e];
    const float* xr = x + (size_t)s * ldx;
    float* tr = tx + (size_t)d * ldt;
    #pragma unroll
    for (int i = 0; i < 4; ++i) {
        int c = c0 + i;
        if (c < Cin) atomicAdd(&tr[c], w * xr[c]);
    }
}

// ---------------- dense part: out = act(x@W0 + tx1@W1 + b) -------------------
// V_WMMA_F32_16X16X4_F32; one wave computes a 64x16 output stripe (4 M-tiles),
// so each B fetch feeds 4 WMMAs. 4 waves/block along N share the 64 A-rows.
// A (16x4 f32): lane l holds row (l&15); v.x = K=2*(l>>4), v.y = K+1.
// B (4x16 f32): lane l holds col (l&15); pair loaded from packed Wp (b64).
// C/D (16x16 f32): VGPR r -> row r + 8*(l>>4), col (l&15).
// avgY != nullptr fuses the residual combine: out = (avgY + relu(gemm+b)) * 0.5
__global__ void k_gemm_wmma(const float* __restrict__ x, int ldx,
                            const float* __restrict__ tx, int ldt,
                            const float* __restrict__ W0p,
                            const float* __restrict__ W1p,
                            const float* __restrict__ bias,
                            float* __restrict__ out, int ldo,
                            int Cin, int Cout, int relu,
                            const float* __restrict__ avgY) {
    const int lane = threadIdx.x & 31;
    const int half = lane >> 4;
    const int lid  = lane & 15;
    const int ntile = blockIdx.y * blockDim.y + threadIdx.y;
    if (ntile * 16 >= Cout) return;            // wave-uniform exit
    const int m0 = blockIdx.x * 64;
    const int n  = ntile * 16 + lid;

    const float* xr0 = x  + (size_t)(m0 + lid) * ldx;
    const float* tr0 = tx + (size_t)(m0 + lid) * ldt;
    const size_t xstep = (size_t)16 * ldx;
    const size_t tstep = (size_t)16 * ldt;
    // packed B: this lane's pair for k-group k0 lives at (((k0>>1)+half)*Cout + n)*2
    const float* bp0 = W0p + ((size_t)half * Cout + n) * 2;
    const float* bp1 = W1p + ((size_t)half * Cout + n) * 2;
    const size_t bstep = (size_t)2 * Cout * 2;   // advance 2 pairs (K+=4) per step

    v8f acc[4] = {};
    const int kmain = Cin & ~3;
    int k0 = 0;
    for (; k0 < kmain; k0 += 4) {
        const int ka = k0 + (half << 1);
        const size_t boff = (size_t)(k0 >> 1) * ((size_t)Cout * 2);
        v2f b0, b1;
        b0.x = bp0[boff];     b0.y = bp0[boff + 1];
        b1.x = bp1[boff];     b1.y = bp1[boff + 1];
        v2f a[4], t[4];
        #pragma unroll
        for (int mt = 0; mt < 4; ++mt) {
            const float* xr = xr0 + (size_t)mt * xstep;
            const float* tr = tr0 + (size_t)mt * tstep;
            a[mt].x = xr[ka]; a[mt].y = xr[ka + 1];
            t[mt].x = tr[ka]; t[mt].y = tr[ka + 1];
        }
        #pragma unroll
        for (int mt = 0; mt < 4; ++mt) {
            acc[mt] = __builtin_amdgcn_wmma_f32_16x16x4_f32(false, a[mt], false, b0, (short)0, acc[mt], false, false);
            acc[mt] = __builtin_amdgcn_wmma_f32_16x16x4_f32(false, t[mt], false, b1, (short)0, acc[mt], false, false);
        }
    }
    if (k0 < Cin) {  // K tail (only Cin=3 hits this); packed B is zero-padded,
                     // only A needs masking. EXEC reconverges before the WMMAs.
        const int ka = k0 + (half << 1);
        const size_t boff = (size_t)(k0 >> 1) * ((size_t)Cout * 2);
        v2f b0, b1;
        b0.x = bp0[boff];     b0.y = bp0[boff + 1];
        b1.x = bp1[boff];     b1.y = bp1[boff + 1];
        v2f a[4] = {}, t[4] = {};
        if (ka < Cin) {
            #pragma unroll
            for (int mt = 0; mt < 4; ++mt) {
                a[mt].x = xr0[(size_t)mt * xstep + ka];
                t[mt].x = tr0[(size_t)mt * tstep + ka];
            }
        }
        if (ka + 1 < Cin) {
            #pragma unroll
            for (int mt = 0; mt < 4; ++mt) {
                a[mt].y = xr0[(size_t)mt * xstep + ka + 1];
                t[mt].y = tr0[(size_t)mt * tstep + ka + 1];
            }
        }
        #pragma unroll
        for (int mt = 0; mt < 4; ++mt) {
            acc[mt] = __builtin_amdgcn_wmma_f32_16x16x4_f32(false, a[mt], false, b0, (short)0, acc[mt], false, false);
            acc[mt] = __builtin_amdgcn_wmma_f32_16x16x4_f32(false, t[mt], false, b1, (short)0, acc[mt], false, false);
        }
    }

    const float bv = bias[n];
    #pragma unroll
    for (int mt = 0; mt < 4; ++mt) {
        #pragma unroll
        for (int r = 0; r < 8; ++r) {
            size_t idx = (size_t)(m0 + 16 * mt + r + (half << 3)) * ldo + n;
            float v = acc[mt][r] + bv;
            if (relu) v = fmaxf(v, 0.f);
            if (avgY) v = (avgY[idx] + v) * 0.5f;
            out[idx] = v;
        }
    }
}

// ---------------- Cout==3 cheb (final layer of each block) ----------------
__global__ void k_cheb_small(const float* __restrict__ x, int ldx,
                             const float* __restrict__ tx, int ldt,
                             const float* __restrict__ W0, const float* __restrict__ W1,
                             const float* __restrict__ bias,
                             float* __restrict__ o, int Cin, int V) {
    int t = blockIdx.x * blockDim.x + threadIdx.x;
    if (t >= V * 3) return;
    int v = t / 3, c = t - v * 3;
    const float* xr = x + (size_t)v * ldx;
    const float* tr = tx + (size_t)v * ldt;
    float acc = bias[c];
    for (int k = 0; k < Cin; ++k)
        acc = fmaf(xr[k], W0[k * 3 + c], fmaf(tr[k], W1[k * 3 + c], acc));
    o[t] = acc;
}

// ---------------- trilinear sampling (grid_sample, align_corners=False, zeros pad) ----
__global__ void k_trilerp(const float* __restrict__ vol, int C, int S,
                          const float* __restrict__ mesh,
                          float* __restrict__ o, int ldo, int V) {
    long t = (long)blockIdx.x * blockDim.x + threadIdx.x;
    if (t >= (long)V * C) return;
    int v = (int)(t / C), c = (int)(t - (long)v * C);
    float fx = ((mesh[v * 3 + 0] + 1.f) * S - 1.f) * 0.5f;
    float fy = ((mesh[v * 3 + 1] + 1.f) * S - 1.f) * 0.5f;
    float fz = ((mesh[v * 3 + 2] + 1.f) * S - 1.f) * 0.5f;
    float x0 = floorf(fx), y0 = floorf(fy), z0 = floorf(fz);
    const float* volc = vol + (size_t)c * S * S * S;
    float acc = 0.f;
    #pragma unroll
    for (int dz = 0; dz < 2; ++dz)
    #pragma unroll
    for (int dy = 0; dy < 2; ++dy)
    #pragma unroll
    for (int dx = 0; dx < 2; ++dx) {
        float xi = x0 + dx, yi = y0 + dy, zi = z0 + dz;
        float w = (1.f - fabsf(fx - xi)) * (1.f - fabsf(fy - yi)) * (1.f - fabsf(fz - zi));
        bool valid = (xi >= 0.f) && (xi < (float)S) &&
                     (yi >= 0.f) && (yi < (float)S) &&
                     (zi >= 0.f) && (zi < (float)S);
        int xii = (int)xi; xii = xii < 0 ? 0 : (xii > S - 1 ? S - 1 : xii);
        int yii = (int)yi; yii = yii < 0 ? 0 : (yii > S - 1 ? S - 1 : yii);
        int zii = (int)zi; zii = zii < 0 ? 0 : (zii > S - 1 ? S - 1 : zii);
        acc += valid ? w * volc[((size_t)zii * S + yii) * S + xii] : 0.f;
    }
    o[(size_t)v * ldo + c] = acc;
}

// ---------------- elementwise glue ----------------
__global__ void k_meshupd(float* __restrict__ mesh, const float* __restrict__ res,
                          float* __restrict__ outsec, int n) {
    int t = blockIdx.x * blockDim.x + threadIdx.x;
    if (t >= n) return;
    float m = mesh[t] + 0.1f * res[t];
    mesh[t] = m;
    outsec[t] = m;
}

// ---------------- host orchestration ----------------
extern "C" void kernel_launch(void* const* d_in, const int* in_sizes, int n_in,
                              void* d_out, int out_size, void* d_ws, size_t ws_size,
                              hipStream_t stream) {
    (void)in_sizes; (void)n_in; (void)out_size; (void)ws_size;
    const int V = V_N, E = E_N, VP = VP_N;

    const float* fvol[5];
    for (int i = 0; i < 5; ++i) fvol[i] = (const float*)d_in[i];
    static const int FCc[5] = {16, 32, 64, 128, 256};
    static const int SZs[5] = {96, 48, 24, 12, 6};
    const float* meshT = (const float*)d_in[5];
    const int* srcI = (const int*)d_in[6];
    const int* dstI = srcI + E;
    float* out = (float*)d_out;

    // params pytree flatten order (jax: dict keys sorted -> "blocks","init"):
    // block b base: 7 + 24*b ; within block: final W0,W1,b @ +0..2 ; init @ +3..5 ;
    //   res pair j cheb t @ +6 + 6*j + 3*t ; top-level init list base = 79.
    auto W = [&](int i) { return (const float*)d_in[i]; };
    const int blockBase[3] = {7, 31, 55};
    const int initBase = 79;

    // workspace carve-up (floats)
    float* p = (float*)d_ws;
    float* deg = p;      p += V;
    float* nrm = p;      p += E;
    float* mesh = p;     p += V * 3;
    float* meshPad = p;  p += VP * 3;
    float* cat = p;      p += (size_t)VP * 768;
    float* yb = p;       p += (size_t)VP * 288;
    float* rb = p;       p += (size_t)VP * 288;
    float* tx = p;       p += (size_t)VP * 768;
    float* res3 = p;     p += V * 3;
    float* wp0 = p;      p += 225000;   // max packed W: 384 pairs * 288 cols * 2
    float* wp1 = p;      p += 225000;

    // graph Laplacian normalization
    hipMemsetAsync(deg, 0, V * sizeof(float), stream);
    k_deg<<<(E + 255) / 256, 256, 0, stream>>>(srcI, deg, E);
    k_norm<<<(E + 255) / 256, 256, 0, stream>>>(srcI, dstI, deg, nrm, E);
    k_initmesh<<<(VP * 3 + 255) / 256, 256, 0, stream>>>(meshT, mesh, meshPad, V, VP);

    auto cheb_big = [&](const float* x, int ldx, int Cin, int Cout,
                        const float* W0, const float* W1, const float* bias,
                        float* o, int ldo, int relu, const float* avgY) {
        hipMemsetAsync(tx, 0, (size_t)VP * Cin * sizeof(float), stream);
        int nC4 = (Cin + 3) / 4;
        long tot = (long)E * nC4;
        k_scatter<<<(unsigned)((tot + 255) / 256), 256, 0, stream>>>(
            srcI, dstI, nrm, x, ldx, tx, Cin, Cin, E, nC4);
        int nQ = nC4 * 2;  // packed pairs, K zero-padded to multiple of 4
        k_packW<<<(nQ * Cout + 255) / 256, 256, 0, stream>>>(W0, wp0, Cin, Cout, nQ);
        k_packW<<<(nQ * Cout + 255) / 256, 256, 0, stream>>>(W1, wp1, Cin, Cout, nQ);
        int ntiles = Cout / 16;
        dim3 g(VP / 64, (ntiles + 3) / 4);
        dim3 b(32, 4);
        k_gemm_wmma<<<g, b, 0, stream>>>(x, ldx, tx, Cin, wp0, wp1, bias, o, ldo,
                                         Cin, Cout, relu, avgY);
    };
    auto cheb_final3 = [&](const float* x, int ldx, int Cin,
                           const float* W0, const float* W1, const float* bias, float* o) {
        hipMemsetAsync(tx, 0, (size_t)VP * Cin * sizeof(float), stream);
        int nC4 = (Cin + 3) / 4;
        long tot = (long)E * nC4;
        k_scatter<<<(unsigned)((tot + 255) / 256), 256, 0, stream>>>(
            srcI, dstI, nrm, x, ldx, tx, Cin, Cin, E, nC4);
        k_cheb_small<<<(V * 3 + 255) / 256, 256, 0, stream>>>(x, ldx, tx, Cin, W0, W1, bias, o, Cin, V);
    };

    const int Cinit[3] = {384, 144, 64};
    const int Ccat[3]  = {768, 240, 112};
    const int hid[3]   = {288, 96, 32};
    const int lv0[3] = {3, 1, 0}, lv1[3] = {4, 2, 1};

    const float* x = meshPad;
    int ldx = 3, cinPrev = 3;
    for (int i = 0; i < 3; ++i) {
        const int ib = initBase + 3 * i;
        // level init cheb + relu -> cat[:, 0:Cinit]
        cheb_big(x, ldx, cinPrev, Cinit[i], W(ib), W(ib + 1), W(ib + 2), cat, Ccat[i], 1, nullptr);
        // trilinear projections -> cat[:, Cinit:Ccat]
        int l0 = lv0[i], l1 = lv1[i];
        k_trilerp<<<(unsigned)(((long)V * FCc[l0] + 255) / 256), 256, 0, stream>>>(
            fvol[l0], FCc[l0], SZs[l0], mesh, cat + Cinit[i], Ccat[i], V);
        k_trilerp<<<(unsigned)(((long)V * FCc[l1] + 255) / 256), 256, 0, stream>>>(
            fvol[l1], FCc[l1], SZs[l1], mesh, cat + Cinit[i] + FCc[l0], Ccat[i], V);
        // gc_block
        const int bb = blockBase[i];
        cheb_big(cat, Ccat[i], Ccat[i], hid[i], W(bb + 3), W(bb + 4), W(bb + 5), yb, hid[i], 1, nullptr);
        for (int j = 0; j < 3; ++j) {
            const int r0 = bb + 6 + 6 * j, r1 = r0 + 3;
            // r = relu(cheb(y))
            cheb_big(yb, hid[i], hid[i], hid[i], W(r0), W(r0 + 1), W(r0 + 2), rb, hid[i], 1, nullptr);
            // y = (y + relu(cheb(r))) * 0.5   (residual average fused in epilogue)
            cheb_big(rb, hid[i], hid[i], hid[i], W(r1), W(r1 + 1), W(r1 + 2), yb, hid[i], 1, yb);
        }
        cheb_final3(yb, hid[i], hid[i], W(bb), W(bb + 1), W(bb + 2), res3);
        k_meshupd<<<(V * 3 + 255) / 256, 256, 0, stream>>>(mesh, res3, out + (size_t)i * V * 3, V * 3);
        x = yb; ldx = hid[i]; cinPrev = hid[i];
    }
}